// MyGraphConv_19361712570525
// MI455X (gfx1250) — compile-verified
//
#include <hip/hip_runtime.h>
#include <math.h>

#define D 64            // D_IN == D_OUT == 64
#define COL_TILES (D / 16)

typedef __attribute__((ext_vector_type(2))) float v2f;
typedef __attribute__((ext_vector_type(8))) float v8f;

__device__ __forceinline__ void atomic_add_f32(float* p, float v) {
  (void)__hip_atomic_fetch_add(p, v, __ATOMIC_RELAXED, __HIP_MEMORY_SCOPE_AGENT);
}

// ---- degree = 1 + #offdiag edges (row side) -------------------------------
__global__ void gcn_deg_init(float* __restrict__ deg, int n) {
  int i = blockIdx.x * blockDim.x + threadIdx.x;
  if (i < n) deg[i] = 1.0f;
}

__global__ void gcn_deg_accum(const long long* __restrict__ row,
                              const long long* __restrict__ col,
                              float* __restrict__ deg, int E) {
  int e = blockIdx.x * blockDim.x + threadIdx.x;
  if (e >= E) return;
  long long r = row[e], c = col[e];
  if (r != c) atomic_add_f32(&deg[r], 1.0f);
}

__global__ void gcn_dinv(const float* __restrict__ deg, float* __restrict__ dinv, int n) {
  int i = blockIdx.x * blockDim.x + threadIdx.x;
  if (i < n) dinv[i] = 1.0f / sqrtf(deg[i]);
}

// ---- per-edge normalized weight (0 for diagonal edges) --------------------
__global__ void gcn_edge_w(const long long* __restrict__ row,
                           const long long* __restrict__ col,
                           const float* __restrict__ dinv,
                           float* __restrict__ ew, int E) {
  int e = blockIdx.x * blockDim.x + threadIdx.x;
  if (e >= E) return;
  long long r = row[e], c = col[e];
  ew[e] = (r != c) ? dinv[r] * dinv[c] : 0.0f;
}

// ---- support = x @ W via V_WMMA_F32_16X16X4_F32 ---------------------------
// One wave per 16x16 output tile; K=64 in 16 steps of 4.
// A (16x4 f32): lanes 0-15 hold (M=lane, K=kk+{0,1}); lanes 16-31 (M=lane-16, K=kk+{2,3})
// B (4x16 f32): symmetric with N across lanes.
// C/D (16x16 f32): VGPR v, lanes 0-15 -> (M=v, N=lane); lanes 16-31 -> (M=v+8, N=lane-16)
__global__ void gcn_gemm_wmma(const float* __restrict__ x,
                              const float* __restrict__ w,
                              float* __restrict__ support,
                              int nTiles) {
  int wid = (blockIdx.x * blockDim.x + threadIdx.x) >> 5;
  if (wid >= nTiles) return;                 // wave-uniform: EXEC stays all-1s
  int lane = threadIdx.x & 31;
  int mt = wid / COL_TILES;
  int nt = wid - mt * COL_TILES;
  int m0 = mt << 4, n0 = nt << 4;
  int half = lane >> 4;                      // 0 or 1
  int l = lane & 15;

  const float* arow = x + (size_t)(m0 + l) * D + 2 * half;   // + k -> x[m0+l][k+2*half]
  const float* bcol = w + n0 + l;                            // + k*D -> W[k][n0+l]

  v8f acc = {0.f, 0.f, 0.f, 0.f, 0.f, 0.f, 0.f, 0.f};
#pragma unroll
  for (int k = 0; k < D; k += 4) {
    v2f a = *(const v2f*)(arow + k);         // 8B aligned: even float offset
    int kb = k + 2 * half;
    v2f b;
    b.x = bcol[(size_t)kb * D];
    b.y = bcol[(size_t)(kb + 1) * D];
    acc = __builtin_amdgcn_wmma_f32_16x16x4_f32(
        /*neg_a=*/false, a, /*neg_b=*/false, b,
        /*c_mod=*/(short)0, acc, /*reuse_a=*/false, /*reuse_b=*/false);
  }

  float* ocol = support + (size_t)n0 + l;
  int mbase = m0 + half * 8;
#pragma unroll
  for (int v = 0; v < 8; ++v)
    ocol[(size_t)(mbase + v) * D] = acc[v];
}

// ---- out = bias + dinv^2 * support (writes every element; no memset needed)
__global__ void gcn_self_bias(const float* __restrict__ support,
                              const float* __restrict__ dinv,
                              const float* __restrict__ bias,
                              float* __restrict__ out, int n_elem) {
  int i = blockIdx.x * blockDim.x + threadIdx.x;
  if (i >= n_elem) return;
  int node = i >> 6;                         // /64
  int d = i & (D - 1);
  float di = dinv[node];
  out[i] = bias[d] + di * di * support[i];
}

// ---- edge scatter: one wave per edge, float2 per lane, fp32 L2 atomics ----
__global__ void gcn_scatter(const long long* __restrict__ row,
                            const long long* __restrict__ col,
                            const float* __restrict__ ew,
                            const float* __restrict__ support,
                            float* __restrict__ out, int E) {
  int wid = (int)(((size_t)blockIdx.x * blockDim.x + threadIdx.x) >> 5);
  if (wid >= E) return;
  float wgt = ew[wid];
  if (wgt == 0.0f) return;                   // masked diagonal edge
  int lane = threadIdx.x & 31;
  long long r = row[wid], c = col[wid];
  v2f sv = *((const v2f*)(support + (size_t)c * D) + lane);
  float* o = out + (size_t)r * D + 2 * lane;
  atomic_add_f32(o, wgt * sv.x);
  atomic_add_f32(o + 1, wgt * sv.y);
}

extern "C" void kernel_launch(void* const* d_in, const int* in_sizes, int n_in,
                              void* d_out, int out_size, void* d_ws, size_t ws_size,
                              hipStream_t stream) {
  const float* x        = (const float*)d_in[0];
  const long long* adj  = (const long long*)d_in[1];   // int64 [2, E] flat
  const float* weight   = (const float*)d_in[2];
  const float* bias     = (const float*)d_in[3];
  float* out            = (float*)d_out;

  const int N = in_sizes[0] / D;     // 50000
  const int E = in_sizes[1] / 2;     // 800000
  const long long* rowp = adj;
  const long long* colp = adj + E;

  // workspace carve-out, 256B aligned slices
  char* ws = (char*)d_ws;
  size_t off = 0;
  auto take = [&](size_t bytes) -> char* {
    char* p = ws + off;
    off = (off + bytes + 255) & ~(size_t)255;
    return p;
  };
  float* deg     = (float*)take((size_t)N * sizeof(float));
  float* dinv    = (float*)take((size_t)N * sizeof(float));
  float* ew      = (float*)take((size_t)E * sizeof(float));
  float* support = (float*)take((size_t)N * D * sizeof(float));
  (void)ws_size;

  const int T = 256;
  gcn_deg_init<<<(N + T - 1) / T, T, 0, stream>>>(deg, N);
  gcn_deg_accum<<<(E + T - 1) / T, T, 0, stream>>>(rowp, colp, deg, E);
  gcn_dinv<<<(N + T - 1) / T, T, 0, stream>>>(deg, dinv, N);
  gcn_edge_w<<<(E + T - 1) / T, T, 0, stream>>>(rowp, colp, dinv, ew, E);

  const int nTiles = ((N + 15) / 16) * COL_TILES;        // 3125 * 4 = 12500 waves
  gcn_gemm_wmma<<<(nTiles + 7) / 8, 256, 0, stream>>>(x, weight, support, nTiles);

  const int nElem = N * D;
  gcn_self_bias<<<(nElem + T - 1) / T, T, 0, stream>>>(support, dinv, bias, out, nElem);

  gcn_scatter<<<(E + 7) / 8, 256, 0, stream>>>(rowp, colp, ew, support, out, E);
}